// SegGPS_66949950210076
// MI455X (gfx1250) — compile-verified
//
#include <hip/hip_runtime.h>

// Final m-reduction: D = A(16x4,=ones) * B(4x16) with V_WMMA_F32_16X16X4_F32,
// so D[i,j] = sum_k B[k,j] = column sums of the wave's 64 partial products
// (2 per lane = exactly the 2-VGPR B operand). A=ones makes this robust to the
// exact B slot->(K,N) bijection; a 4-step shfl butterfly over 16 lanes finishes.
typedef __attribute__((ext_vector_type(2))) float v2f;
typedef __attribute__((ext_vector_type(8))) float v8f;

static constexpr int  L_SITES = 64;
static constexpr int  M_DIM   = 64;
static constexpr int  UPD     = 65;                    // MAX_UP+1 == MAX_DN+1
static constexpr int  S_L     = UPD * UPD;             // 4225     site stride (elems)
static constexpr int  S_M     = L_SITES * S_L;         // 270400   m stride (elems)
static constexpr long S_D     = (long)M_DIM * S_M;     // 17305600 local-dim stride

__global__ __launch_bounds__(256)
void seg_gps_kernel(const int* __restrict__ idx,
                    const float* __restrict__ eps,
                    float* __restrict__ out,
                    int nB)
{
    const int lane = threadIdx.x & 31;
    const int wave = threadIdx.x >> 5;
    const int b    = blockIdx.x * 8 + wave;
    if (b >= nB) return;                 // whole wave exits together: EXEC stays all-1s

    // Coalesced load of this batch's 64 site indices: 2 per lane.
    const int* bi   = idx + (size_t)b * L_SITES;
    const int  d_lo = bi[lane];          // sites 0..31
    const int  d_hi = bi[lane + 32];     // sites 32..63

    // Loop-invariant per-lane gather offsets (elements): m = lane and m = lane+32.
    // Max byte offset = 63*270400*4 = 68.1 MB < 2^31 -> fits the 32-bit voffset
    // of the saddr+voffset global-load form; the per-site base stays in SGPRs.
    const int offA = lane * S_M;
    const int offB = (lane + 32) * S_M;

    float p0 = 1.0f, p1 = 1.0f;
    int nup = 0, ndn = 0;                // exclusive prefix counters (SGPR chains)

#define SITE(l, dreg, srclane)                                                  \
    {                                                                           \
        const int d = __builtin_amdgcn_readlane((dreg), (srclane)); /* SGPR */  \
        const float* base = eps + (size_t)d * S_D + (size_t)((l) * S_L)         \
                                + (size_t)(nup * UPD + ndn);                    \
        p0 *= base[offA];                                                       \
        p1 *= base[offB];                                                       \
        nup += d & 1;                                                           \
        ndn += (d >> 1) & 1;                                                    \
    }

    #pragma unroll
    for (int l = 0; l < 32; ++l) SITE(l, d_lo, l);
    #pragma unroll
    for (int l = 32; l < 64; ++l) SITE(l, d_hi, l - 32);
#undef SITE

    // Sum over all 64 m-values with one f32 WMMA: ones(16x4) x B(4x16).
    v2f av; av[0] = 1.0f; av[1] = 1.0f;      // A = all ones (layout-agnostic)
    v2f bv; bv[0] = p0;   bv[1] = p1;        // B = the wave's 64 partial products
    v8f cv = {};
    cv = __builtin_amdgcn_wmma_f32_16x16x4_f32(
            /*neg_a=*/false, av, /*neg_b=*/false, bv,
            /*c_mod=*/(short)0, cv, /*reuse_a=*/false, /*reuse_b=*/false);

    // D rows are identical; VGPR0 holds D[row, col=lane%16]. Sum the 16 columns.
    float t = cv[0];
    t += __shfl_xor(t, 1, 32);
    t += __shfl_xor(t, 2, 32);
    t += __shfl_xor(t, 4, 32);
    t += __shfl_xor(t, 8, 32);
    if (lane == 0) out[b] = t;
}

extern "C" void kernel_launch(void* const* d_in, const int* in_sizes, int n_in,
                              void* d_out, int out_size, void* d_ws, size_t ws_size,
                              hipStream_t stream)
{
    const int*   idx = (const int*)d_in[0];     // (B, 64) int32 (JAX x64 disabled)
    const float* eps = (const float*)d_in[1];   // (4, 64, 64, 65, 65) f32
    float*       out = (float*)d_out;           // (B,) f32

    const int nB     = in_sizes[0] / L_SITES;   // 8192
    const int blocks = (nB + 7) / 8;            // 8 waves (batches) per 256-thread block

    seg_gps_kernel<<<blocks, 256, 0, stream>>>(idx, eps, out, nB);
}